// GAT_ATTLOC_58471684768234
// MI455X (gfx1250) — compile-verified
//
#include <hip/hip_runtime.h>
#include <hip/hip_bf16.h>

#define NN 50000
#define EE 1600000
#define KDIM 256
#define FOUT 64
#define NEG_SLOPE 0.2f
#define EPSI 1e-16f

typedef float v2f __attribute__((ext_vector_type(2)));
typedef float v8f __attribute__((ext_vector_type(8)));

// Order-preserving float <-> uint encoding so atomicMax(uint) == float max.
__device__ __forceinline__ unsigned fenc(float f) {
    unsigned u = __float_as_uint(f);
    return ((int)u < 0) ? ~u : (u | 0x80000000u);
}
__device__ __forceinline__ float fdec(unsigned e) {
    return ((int)e < 0) ? __uint_as_float(e & 0x7fffffffu) : __uint_as_float(~e);
}

// ---------------- init: zero agg / denom / m_enc ----------------
__global__ void k_zero(float* __restrict__ p, int n) {
    int i = blockIdx.x * blockDim.x + threadIdx.x;
    int stride = gridDim.x * blockDim.x;
    for (; i < n; i += stride) p[i] = 0.0f;
}

// ---------------- combined attention vectors ----------------
// wsc[k] = sum_f v_mapping[f][k] * att_src[f] / max(||att_src||,1e-12)  (and dst)
__global__ void k_att(const float* __restrict__ vmap,
                      const float* __restrict__ att_s,
                      const float* __restrict__ att_d,
                      float* __restrict__ wsc, float* __restrict__ wdc) {
    float ns = 0.f, nd = 0.f;
    for (int f = 0; f < FOUT; ++f) { ns += att_s[f] * att_s[f]; nd += att_d[f] * att_d[f]; }
    ns = fmaxf(sqrtf(ns), 1e-12f);
    nd = fmaxf(sqrtf(nd), 1e-12f);
    int k = threadIdx.x;  // blockDim.x == 256 == KDIM
    float as = 0.f, ad = 0.f;
    for (int f = 0; f < FOUT; ++f) {
        float v = vmap[f * KDIM + k];
        as += v * att_s[f];
        ad += v * att_d[f];
    }
    wsc[k] = as / ns;
    wdc[k] = ad / nd;
}

// ---------------- WMMA GEMM: x_proj = x @ W  (+ fused score dots) ----------------
// 8 waves/block; wave => one 16-row tile, 4 col tiles of 16.
// V_WMMA_F32_16X16X4_F32 fragment layout (wave32):
//   A 16x4 : lane(l<16): M=l, v2f = {A[M][kb+0],A[M][kb+1]}; lane l+16: {A[M][kb+2],A[M][kb+3]}
//   B 4x16 : lane(l<16): N=l, v2f = {B[kb+0][N],B[kb+1][N]}; lane l+16: {B[kb+2][N],B[kb+3][N]}
//   C/D    : vgpr j, lanes0-15: (M=j, N=lane), lanes16-31: (M=j+8, N=lane-16)
__global__ __launch_bounds__(256)
void k_gemm(const float* __restrict__ x, const float* __restrict__ W,
            const float* __restrict__ wsc, const float* __restrict__ wdc,
            float* __restrict__ xp, float* __restrict__ asrc, float* __restrict__ adst) {
    // W transposed into pairs along K, XOR-swizzled to kill LDS bank conflicts. 64KB.
    __shared__ v2f WtL[FOUT * 128];
    for (int idx = threadIdx.x; idx < FOUT * 128; idx += 256) {
        int n = idx & 63;        // output column (coalesced global reads)
        int p = idx >> 6;        // k-pair index, k = 2p
        v2f v;
        v.x = W[(2 * p) * FOUT + n];
        v.y = W[(2 * p + 1) * FOUT + n];
        WtL[(n << 7) | (p ^ (n & 15))] = v;
    }
    __syncthreads();

    int lane = threadIdx.x & 31;
    int wv   = threadIdx.x >> 5;
    int tile = blockIdx.x * 8 + wv;
    if (tile >= NN / 16) return;  // NN % 16 == 0, no row tail
    int r0  = tile * 16;
    int mlo = lane & 15;
    int hi  = lane >> 4;

    v8f c0 = {}, c1 = {}, c2 = {}, c3 = {};
    float ss = 0.f, sd = 0.f;
    const float* xrow = x + (size_t)(r0 + mlo) * KDIM;

#pragma unroll 4
    for (int s = 0; s < 64; ++s) {
        int k = 4 * s + hi * 2;                    // even => 8B aligned
        v2f a  = *(const v2f*)(xrow + k);          // A fragment (global, b64)
        v2f w0 = *(const v2f*)(wsc + k);           // fused score dots
        v2f w1 = *(const v2f*)(wdc + k);
        ss += a.x * w0.x + a.y * w0.y;
        sd += a.x * w1.x + a.y * w1.y;
        int p = k >> 1;
        v2f b0 = WtL[(( 0 + mlo) << 7) | (p ^ mlo)];
        v2f b1 = WtL[((16 + mlo) << 7) | (p ^ mlo)];
        v2f b2 = WtL[((32 + mlo) << 7) | (p ^ mlo)];
        v2f b3 = WtL[((48 + mlo) << 7) | (p ^ mlo)];
        c0 = __builtin_amdgcn_wmma_f32_16x16x4_f32(false, a, false, b0, (short)0, c0, false, false);
        c1 = __builtin_amdgcn_wmma_f32_16x16x4_f32(false, a, false, b1, (short)0, c1, false, false);
        c2 = __builtin_amdgcn_wmma_f32_16x16x4_f32(false, a, false, b2, (short)0, c2, false, false);
        c3 = __builtin_amdgcn_wmma_f32_16x16x4_f32(false, a, false, b3, (short)0, c3, false, false);
    }

    // lane l and l+16 hold complementary halves of row l's dot products
    float st = ss + __shfl_xor(ss, 16);
    float dt = sd + __shfl_xor(sd, 16);
    if (hi == 0) { asrc[r0 + mlo] = st; adst[r0 + mlo] = dt; }

#pragma unroll
    for (int j = 0; j < 8; ++j) {
        int row = r0 + j + 8 * hi;
        float* o = xp + (size_t)row * FOUT + mlo;
        o[0]  = c0[j];
        o[16] = c1[j];
        o[32] = c2[j];
        o[48] = c3[j];
    }
}

// ---------------- edge pass 1: raw alpha + segment max ----------------
__global__ void k_edge_max(const int* __restrict__ ei,
                           const float* __restrict__ asrc, const float* __restrict__ adst,
                           float* __restrict__ aw, unsigned* __restrict__ menc) {
    int e = blockIdx.x * blockDim.x + threadIdx.x;
    if (e >= EE) return;
    int s = ei[e], d = ei[EE + e];
    float a = asrc[s] + adst[d];
    a = (a > 0.f) ? a : NEG_SLOPE * a;  // leaky relu
    aw[e] = a;
    atomicMax(&menc[d], fenc(a));
}

// ---------------- edge pass 2: exp + denom (in-place w) ----------------
__global__ void k_edge_exp(const int* __restrict__ ei, float* __restrict__ aw,
                           const unsigned* __restrict__ menc, float* __restrict__ den) {
    int e = blockIdx.x * blockDim.x + threadIdx.x;
    if (e >= EE) return;
    int d = ei[EE + e];
    float w = expf(aw[e] - fdec(menc[d]));
    aw[e] = w;
    atomicAdd(&den[d], w);
}

// ---------------- edge pass 3: weighted scatter (unnormalized) ----------------
// 64 threads per edge => coalesced 128B gathers and 128B atomic bursts.
__global__ void k_scatter(const int* __restrict__ ei, const float* __restrict__ aw,
                          const float* __restrict__ xp, float* __restrict__ agg) {
    int tid = blockIdx.x * blockDim.x + threadIdx.x;  // EE*64 = 102.4M < 2^31
    int e = tid >> 6, c = tid & 63;
    int s = ei[e], d = ei[EE + e];
    atomicAdd(&agg[(size_t)d * FOUT + c], aw[e] * xp[(size_t)s * FOUT + c]);
}

// ---------------- final: residual + deferred softmax normalization + bias ----------------
__global__ void k_final(const float* __restrict__ xp, const float* __restrict__ agg,
                        const float* __restrict__ den, const float* __restrict__ bias,
                        float* __restrict__ out) {
    int tid = blockIdx.x * blockDim.x + threadIdx.x;
    if (tid >= NN * FOUT) return;
    int n = tid >> 6, c = tid & 63;
    out[tid] = xp[tid] + agg[tid] / (den[n] + EPSI) + bias[c];
}

extern "C" void kernel_launch(void* const* d_in, const int* in_sizes, int n_in,
                              void* d_out, int out_size, void* d_ws, size_t ws_size,
                              hipStream_t stream) {
    const float* x    = (const float*)d_in[0];
    const int*   ei   = (const int*)d_in[1];
    const float* vmap = (const float*)d_in[2];
    const float* W    = (const float*)d_in[3];
    const float* atts = (const float*)d_in[4];
    const float* attd = (const float*)d_in[5];
    const float* bias = (const float*)d_in[6];
    float* out = (float*)d_out;

    float* ws = (float*)d_ws;
    float*    xp   = ws;                          // [NN*64]
    float*    agg  = xp + (size_t)NN * FOUT;      // [NN*64]
    float*    den  = agg + (size_t)NN * FOUT;     // [NN]
    unsigned* menc = (unsigned*)(den + NN);       // [NN]
    float*    asrc = (float*)(menc + NN);         // [NN]
    float*    adst = asrc + NN;                   // [NN]
    float*    aw   = adst + NN;                   // [EE]
    float*    wsc  = aw + EE;                     // [256]
    float*    wdc  = wsc + KDIM;                  // [256]

    // agg, den, menc are contiguous: NN*66 dwords, all-zero init each call
    k_zero<<<2048, 256, 0, stream>>>(agg, NN * 66);
    k_att<<<1, 256, 0, stream>>>(vmap, atts, attd, wsc, wdc);
    k_gemm<<<(NN / 16 + 7) / 8, 256, 0, stream>>>(x, W, wsc, wdc, xp, asrc, adst);
    k_edge_max<<<(EE + 255) / 256, 256, 0, stream>>>(ei, asrc, adst, aw, menc);
    k_edge_exp<<<(EE + 255) / 256, 256, 0, stream>>>(ei, aw, menc, den);
    k_scatter<<<(EE * 64) / 256, 256, 0, stream>>>(ei, aw, xp, agg);   // exact multiple
    k_final<<<(NN * FOUT + 255) / 256, 256, 0, stream>>>(xp, agg, den, bias, out);
}